// CandidateAwareClock_74448963109132
// MI455X (gfx1250) — compile-verified
//
#include <hip/hip_runtime.h>
#include <hip/hip_bf16.h>
#include <math.h>

#define BB 4096
#define LL 200
#define DD 64
#define NT 13   // ceil(200/16) position tiles per row

typedef float v2f __attribute__((ext_vector_type(2)));
typedef float v8f __attribute__((ext_vector_type(8)));

__global__ __launch_bounds__(256, 1)
void cand_attn_kernel(const int* __restrict__ items_pad,
                      const int* __restrict__ dts_pad,
                      const unsigned char* __restrict__ mask,
                      const int* __restrict__ cand_items,
                      const float* __restrict__ item_emb,
                      const float* __restrict__ dt_gate,
                      const float* __restrict__ raw_tau,
                      float* __restrict__ out_u,
                      float* __restrict__ out_attn,
                      float* __restrict__ out_tau)
{
    __shared__ float q_s[DD];
    __shared__ int   items_s[NT * 16];
    __shared__ float sims_s[NT * 16];
    __shared__ float attn_s[LL];
    __shared__ float red_s[8];
    __shared__ float part_s[4 * DD];
    __shared__ float bcast_s;

    const int b    = blockIdx.x;
    const int tid  = threadIdx.x;
    const int lane = tid & 31;
    const int wave = tid >> 5;

    // tau = softplus(raw_tau) + 1e-6 (computed redundantly; one scalar load)
    const float rt  = raw_tau[0];
    const float tau = fmaxf(rt, 0.0f) + log1pf(__expf(-fabsf(rt))) + 1e-6f;

    // ---- Phase 1: stage q and the row's item indices in LDS ----
    if (tid < DD) q_s[tid] = item_emb[cand_items[b] * DD + tid];
    for (int i = tid; i < NT * 16; i += 256) {
        const int p = (i < LL) ? i : (LL - 1);   // clamp pad-tile positions
        items_s[i] = items_pad[b * LL + p];
    }
    __syncthreads();

    // ---- Phase 2: sim = K . q via fp32 WMMA (16 positions / tile) ----
    // B-matrix (4x16): q chunk broadcast to all 16 columns.
    // Layout mirrors A: lanes 0-15 hold rows K=0,1; lanes 16-31 hold K=2,3.
    const int half = (lane >= 16) ? 2 : 0;
    v2f qb[16];
    #pragma unroll
    for (int j = 0; j < 16; ++j) {
        qb[j].x = q_s[4 * j + half];
        qb[j].y = q_s[4 * j + half + 1];
    }

    for (int t = wave; t < NT; t += 8) {          // wave-uniform: EXEC stays all-1s
        const int m   = lane & 15;
        const int row = items_s[t * 16 + m];
        const v2f* kr2 = (const v2f*)(item_emb + (long)row * DD);
        v2f a[16];
        #pragma unroll
        for (int j = 0; j < 16; ++j)               // aligned global_load_b64 gathers
            a[j] = kr2[2 * j + (half >> 1)];

        v8f acc = {0.f, 0.f, 0.f, 0.f, 0.f, 0.f, 0.f, 0.f};
        #pragma unroll
        for (int j = 0; j < 16; ++j) {
            acc = __builtin_amdgcn_wmma_f32_16x16x4_f32(
                /*neg_a=*/false, a[j], /*neg_b=*/false, qb[j],
                /*c_mod=*/(short)0, acc, /*reuse_a=*/false, /*reuse_b=*/false);
        }
        // C/D layout: VGPR r -> M=r (lanes 0-15), M=8+r (lanes 16-31); all columns equal.
        if (lane == 0) {
            #pragma unroll
            for (int r = 0; r < 8; ++r) sims_s[t * 16 + r] = acc[r];
        }
        if (lane == 16) {
            #pragma unroll
            for (int r = 0; r < 8; ++r) sims_s[t * 16 + 8 + r] = acc[r];
        }
    }
    __syncthreads();

    // ---- Phase 3: gated, masked softmax over L ----
    float logit = -INFINITY;
    if (tid < LL) {
        const int   dt = dts_pad[b * LL + tid];
        const float g  = (dt == 0) ? 0.0f : dt_gate[dt];      // padding_idx=0
        const float s  = sims_s[tid];
        logit = mask[b * LL + tid] ? (s * g / tau) : -INFINITY;
    }

    float v = logit;
    #pragma unroll
    for (int off = 16; off; off >>= 1) v = fmaxf(v, __shfl_xor(v, off, 32));
    if (lane == 0) red_s[wave] = v;
    __syncthreads();
    if (tid == 0) {
        float mx = red_s[0];
        #pragma unroll
        for (int w = 1; w < 8; ++w) mx = fmaxf(mx, red_s[w]);
        bcast_s = mx;
    }
    __syncthreads();
    const float mx = bcast_s;

    const float e = (tid < LL) ? __expf(logit - mx) : 0.0f;
    v = e;
    #pragma unroll
    for (int off = 16; off; off >>= 1) v += __shfl_xor(v, off, 32);
    if (lane == 0) red_s[wave] = v;
    __syncthreads();
    if (tid == 0) {
        float s = 0.0f;
        #pragma unroll
        for (int w = 0; w < 8; ++w) s += red_s[w];
        bcast_s = s;
    }
    __syncthreads();
    const float inv = 1.0f / bcast_s;
    if (tid < LL) {
        const float a = e * inv;
        attn_s[tid] = a;
        out_attn[b * LL + tid] = a;
    }
    __syncthreads();

    // ---- Phase 4: u = attn^T . K (coalesced over d; L2-resident gathers) ----
    const int d   = tid & 63;
    const int grp = tid >> 6;
    float acc = 0.0f;
    for (int l = grp; l < LL; l += 4)
        acc += attn_s[l] * item_emb[(long)items_s[l] * DD + d];
    part_s[grp * DD + d] = acc;
    __syncthreads();
    if (tid < DD) {
        out_u[b * DD + tid] = part_s[tid] + part_s[DD + tid] +
                              part_s[2 * DD + tid] + part_s[3 * DD + tid];
    }
    if (b == 0 && tid == 0) out_tau[0] = tau;
}

extern "C" void kernel_launch(void* const* d_in, const int* in_sizes, int n_in,
                              void* d_out, int out_size, void* d_ws, size_t ws_size,
                              hipStream_t stream) {
    (void)in_sizes; (void)n_in; (void)out_size; (void)d_ws; (void)ws_size;
    const int*           items_pad  = (const int*)d_in[0];
    const int*           dts_pad    = (const int*)d_in[1];
    const unsigned char* mask       = (const unsigned char*)d_in[2]; // jnp bool, 1 byte
    const int*           cand_items = (const int*)d_in[3];
    const float*         item_emb   = (const float*)d_in[4];
    const float*         dt_gate    = (const float*)d_in[5];
    const float*         raw_tau    = (const float*)d_in[6];

    float* out      = (float*)d_out;
    float* out_u    = out;                       // [B, D]
    float* out_attn = out + (size_t)BB * DD;     // [B, L]
    float* out_tau  = out_attn + (size_t)BB * LL;// scalar

    cand_attn_kernel<<<BB, 256, 0, stream>>>(items_pad, dts_pad, mask, cand_items,
                                             item_emb, dt_gate, raw_tau,
                                             out_u, out_attn, out_tau);
}